// Eikonal3D_78022375899315
// MI455X (gfx1250) — compile-verified
//
#include <hip/hip_runtime.h>
#include <stdint.h>

#define MDIM 64
#define VOL (64 * 64 * 64)
#define BIGF 1000000000.0f
#define UINITF 1000.0f
#define NITER 200

// CDNA5 async global->LDS path (gfx1250). Probe round 1 confirmed the builtin
// exists with 4 args: (global int* src, lds int* dst, imm offset, imm cpol).
#if __has_builtin(__builtin_amdgcn_global_load_async_to_lds_b32) && \
    __has_builtin(__builtin_amdgcn_s_wait_asynccnt)
#define HAVE_ASYNC_LDS 1
typedef __attribute__((address_space(1))) int* as1_i32p;
typedef __attribute__((address_space(3))) int* as3_i32p;
#else
#define HAVE_ASYNC_LDS 0
#endif

// ---------------------------------------------------------------- init ------
__global__ __launch_bounds__(256) void eik_init(const int* __restrict__ xs,
                                                const int* __restrict__ ys,
                                                const int* __restrict__ zs,
                                                float* __restrict__ u0) {
    const int idx = blockIdx.x * 256 + threadIdx.x;   // 0 .. VOL-1
    const int x = idx >> 12;
    const int y = (idx >> 6) & 63;
    const int z = idx & 63;
    // Source point is given as integers -> floor==ceil, dist==0 -> u0 = 0 there.
    const int X = *xs, Y = *ys, Z = *zs;
    float u = UINITF;
    if (x == X && y == Y && z == Z) u = 0.0f;
    u0[idx] = u;
}

// --------------------------------------------------------------- sweep ------
// One Jacobi relaxation step:  uout = min(uin, local_eikonal_update(uin)).
// Block = (64 z, 4 y) = 8 wave32s; grid = (16 y-groups, 64 x).
__global__ __launch_bounds__(256) void eik_sweep(const float* __restrict__ uin,
                                                 float* __restrict__ uout,
                                                 const float* __restrict__ f) {
    __shared__ float tile[4][64];

    const int tx  = threadIdx.x;                 // z: 0..63
    const int ty  = threadIdx.y;                 // y sub-row: 0..3
    const int x   = blockIdx.y;                  // 0..63
    const int y   = (blockIdx.x << 2) + ty;      // 0..63
    const int idx = (x << 12) + (y << 6) + tx;

#if HAVE_ASYNC_LDS
    // CDNA5 async data path: DMA the center tile straight into LDS,
    // bypassing the VGPR round trip; tracked by ASYNCcnt.
    __builtin_amdgcn_global_load_async_to_lds_b32(
        (as1_i32p)(uin + idx),
        (as3_i32p)&tile[ty][tx],
        0, 0);
    __builtin_amdgcn_s_wait_asynccnt(0);
    __syncthreads();
    const float c = tile[ty][tx];
#else
    const float c = uin[idx];
    tile[ty][tx] = c;
    __syncthreads();
#endif

    // Neighbors: z and interior-y from LDS; x and y-halo from L2-resident global.
    const float zm = (tx > 0)  ? tile[ty][tx - 1] : BIGF;
    const float zp = (tx < 63) ? tile[ty][tx + 1] : BIGF;
    const float ym = (ty > 0) ? tile[ty - 1][tx] : ((y > 0)  ? uin[idx - 64] : BIGF);
    const float yp = (ty < 3) ? tile[ty + 1][tx] : ((y < 63) ? uin[idx + 64] : BIGF);
    const float xm = (x > 0)  ? uin[idx - 4096] : BIGF;
    const float xp = (x < 63) ? uin[idx + 4096] : BIGF;

    const float ax = fminf(xm, xp);
    const float ay = fminf(ym, yp);
    const float az = fminf(zm, zp);

    // Exact 3-element sort via min/max network.
    const float mn01 = fminf(ax, ay);
    const float mx01 = fmaxf(ax, ay);
    const float a  = fminf(mn01, az);                 // smallest
    const float s2 = fmaxf(mx01, az);                 // largest
    const float b  = fmaxf(mn01, fminf(mx01, az));    // median

    const float fhv = 1.0f / f[idx];                  // H == 1.0
    const float fh2 = fhv * fhv;

    const float u1  = a + fhv;
    const float amb = a - b;
    const float d2  = 2.0f * fh2 - amb * amb;
    const float u2  = 0.5f * (a + b + sqrtf(fmaxf(d2, 0.0f)));
    const float abc = a + b + s2;
    const float d3  = abc * abc - 3.0f * (a * a + b * b + s2 * s2 - fh2);
    const float u3  = (abc + sqrtf(fmaxf(d3, 0.0f))) / 3.0f;

    const float unew = (u1 <= b) ? u1 : ((u2 <= s2) ? u2 : u3);
    uout[idx] = fminf(c, unew);
}

// -------------------------------------------------------------- interp ------
__global__ __launch_bounds__(256) void eik_interp(const float* __restrict__ u,
                                                  const float* __restrict__ gx,
                                                  const float* __restrict__ gy,
                                                  const float* __restrict__ gz,
                                                  float* __restrict__ out,
                                                  int n) {
    const int i = blockIdx.x * 256 + threadIdx.x;
    if (i >= n) return;
    const float X = gx[i], Y = gy[i], Z = gz[i];
    const int x1 = (int)floorf(X), y1 = (int)floorf(Y), z1 = (int)floorf(Z);
    const int x2 = (int)ceilf(X),  y2 = (int)ceilf(Y),  z2 = (int)ceilf(Z);

    #define TAKE(ix, iy, iz) u[((ix) << 12) + ((iy) << 6) + (iz)]
    const float f111 = TAKE(x1, y1, z1);
    const float f211 = TAKE(x2, y1, z1);
    const float f121 = TAKE(x1, y2, z1);
    const float f221 = TAKE(x2, y2, z1);
    const float f112 = TAKE(x1, y1, z2);
    const float f212 = TAKE(x2, y1, z2);
    const float f122 = TAKE(x1, y2, z2);
    const float f222 = TAKE(x2, y2, z2);
    #undef TAKE

    const float x1f = (float)x1, x2f = (float)x2;
    const float y1f = (float)y1, y2f = (float)y2;
    const float z1f = (float)z1, z2f = (float)z2;

    const float t =
        (z2f - Z) * (y2f - Y) * ((x2f - X) * f111 + (X - x1f) * f211) +
        (z2f - Z) * (Y - y1f) * ((x2f - X) * f121 + (X - x1f) * f221) +
        (Z - z1f) * (y2f - Y) * ((x2f - X) * f112 + (X - x1f) * f212) +
        (Z - z1f) * (Y - y1f) * ((x2f - X) * f122 + (X - x1f) * f222);
    out[i] = t;
}

// -------------------------------------------------------------- launch ------
extern "C" void kernel_launch(void* const* d_in, const int* in_sizes, int n_in,
                              void* d_out, int out_size, void* d_ws, size_t ws_size,
                              hipStream_t stream) {
    const float* f  = (const float*)d_in[0];
    const int*   xs = (const int*)d_in[1];
    const int*   ys = (const int*)d_in[2];
    const int*   zs = (const int*)d_in[3];
    const float* gx = (const float*)d_in[4];
    const float* gy = (const float*)d_in[5];
    const float* gz = (const float*)d_in[6];
    float* out = (float*)d_out;

    // Ping-pong buffers: needs 2 * VOL * 4 = 2 MB of workspace.
    float* uA = (float*)d_ws;
    float* uB = uA + VOL;

    eik_init<<<VOL / 256, 256, 0, stream>>>(xs, ys, zs, uA);

    dim3 blk(64, 4, 1);
    dim3 grd(16, 64, 1);
    const float* src = uA;
    float*       dst = uB;
    for (int it = 0; it < NITER; ++it) {
        eik_sweep<<<grd, blk, 0, stream>>>(src, dst, f);
        float* t = (float*)src;
        src = dst;
        dst = t;
    }
    // NITER is even -> final result sits back in uA (== src after last swap).

    const int npts = in_sizes[4];
    eik_interp<<<(npts + 255) / 256, 256, 0, stream>>>(src, gx, gy, gz, out, npts);
}